// RetinaLoss_17514876634219
// MI455X (gfx1250) — compile-verified
//
#include <hip/hip_runtime.h>
#include <hip/hip_bf16.h>
#include <math.h>

#define NCLS    80
#define BATCH   4
#define IMG_W   640.0f
#define IMG_H   640.0f
#define ALPHA_F 0.25f
#define BETA_F  (1.0f / 9.0f)
#define EPS_F   1.0e-4f

typedef __attribute__((ext_vector_type(2))) float v2f;
typedef __attribute__((ext_vector_type(8))) float v8f;

// ---------------------------------------------------------------------------
// Exact f32 wave32 sum via V_WMMA_F32_16X16X4_F32.
// A layout (ISA 7.12.2): lane m (0-15) VGPR0 = A[m][0], VGPR1 = A[m][1];
// lane 16+m VGPR0 = A[m][2], VGPR1 = A[m][3].  With a = {acc, 0} and B = ones:
//   D[m][n] = acc_m + acc_{m+16}  (all n identical).
// D layout: lane L holds D[v][L] (v=VGPR 0..7), lane 16+L holds D[v+8][L].
// Sum of 8 D VGPRs per lane -> half-sums; xor-shuffle by 16 -> full wave sum.
// EXEC must be all ones: callers keep every lane active at the call site.
// ---------------------------------------------------------------------------
__device__ __forceinline__ float wave_sum32(float v) {
    v2f a; a[0] = v;    a[1] = 0.0f;
    v2f b; b[0] = 1.0f; b[1] = 1.0f;
    v8f c = {};
    c = __builtin_amdgcn_wmma_f32_16x16x4_f32(
            /*neg_a=*/false, a, /*neg_b=*/false, b,
            /*c_mod=*/(short)0, c, /*reuse_a=*/false, /*reuse_b=*/false);
    float s = ((c[0] + c[1]) + (c[2] + c[3])) + ((c[4] + c[5]) + (c[6] + c[7]));
    s += __shfl_xor(s, 16, 32);
    return s;
}

// Block-wide sum (WMMA wave reduce + LDS) then one atomicAdd per block.
__device__ __forceinline__ void block_atomic_add(float v, float* dst, float* lds) {
    float s = wave_sum32(v);
    const int lane = threadIdx.x & 31;
    const int wid  = threadIdx.x >> 5;
    if (lane == 0) lds[wid] = s;
    __syncthreads();
    if (threadIdx.x == 0) {
        float t = 0.0f;
        const int nw = blockDim.x >> 5;
        for (int i = 0; i < nw; ++i) t += lds[i];
        atomicAdd(dst, t);
    }
    __syncthreads();
}

// ---------------------------------------------------------------------------
// Kernel 0: zero the 12 per-image accumulators.
// ---------------------------------------------------------------------------
__global__ void k_zero(float* acc) {
    if (threadIdx.x < 12) acc[threadIdx.x] = 0.0f;
}

// ---------------------------------------------------------------------------
// Kernel 1: anchor/GT matching + labels + smooth-L1 reg loss + num_pos.
// grid = (ceil(N/256), B), block = 256.
// ---------------------------------------------------------------------------
__global__ void k_match(const float* __restrict__ anchors,
                        const float* __restrict__ reg_heads,
                        const float* __restrict__ annots,
                        float* __restrict__ labels,
                        float* __restrict__ reg_sum,
                        float* __restrict__ num_pos,
                        int N, int M) {
    const int b = blockIdx.y;
    const int n = blockIdx.x * blockDim.x + threadIdx.x;

    __shared__ float ann[128 * 5];   // M <= 128 annotations (M=20 here)
    __shared__ float red[8];

    const int annN = M * 5;
    for (int i = threadIdx.x; i < annN; i += blockDim.x)
        ann[i] = annots[(size_t)b * annN + i];
    __syncthreads();

    float sl1pos = 0.0f;
    float posf   = 0.0f;
    const bool active = (n < N);

    if (active) {
        const size_t base = ((size_t)b * N + n) * 4;
        const float ax0 = anchors[base + 0];
        const float ay0 = anchors[base + 1];
        const float ax1 = anchors[base + 2];
        const float ay1 = anchors[base + 3];
        const float areaA = (ax1 - ax0) * (ay1 - ay0);

        float best = -1.0f;
        int   bidx = 0;
        bool  any_valid = false;
        #pragma unroll 4
        for (int m = 0; m < M; ++m) {
            const float gx0 = ann[m * 5 + 0];
            const float gy0 = ann[m * 5 + 1];
            const float gx1 = ann[m * 5 + 2];
            const float gy1 = ann[m * 5 + 3];
            const bool  gv  = ann[m * 5 + 4] >= 0.0f;
            any_valid |= gv;
            const float iw = fmaxf(fminf(ax1, gx1) - fmaxf(ax0, gx0), 0.0f);
            const float ih = fmaxf(fminf(ay1, gy1) - fmaxf(ay0, gy0), 0.0f);
            const float inter = iw * ih;
            const float areaG = (gx1 - gx0) * (gy1 - gy0);
            const float uni = fmaxf(areaA + areaG - inter, 1.0e-4f);
            float iou = inter / uni;
            iou = gv ? iou : -1.0f;
            if (iou > best) { best = iou; bidx = m; }   // strict '>' == argmax first-max
        }

        float lab = -1.0f;
        if (best < 0.4f)  lab = 0.0f;
        if (best >= 0.5f) lab = ann[bidx * 5 + 4] + 1.0f;
        if (!any_valid)   lab = -1.0f;
        const bool border = (ax0 > 0.0f) && (ay0 > 0.0f) && (ax1 < IMG_W) && (ay1 < IMG_H);
        if (!border) lab = -1.0f;

        labels[(size_t)b * N + n] = lab;

        if (lab > 0.0f) {
            posf = 1.0f;
            const float gx0 = ann[bidx * 5 + 0];
            const float gy0 = ann[bidx * 5 + 1];
            const float gx1 = ann[bidx * 5 + 2];
            const float gy1 = ann[bidx * 5 + 3];
            const float aw = ax1 - ax0, ah = ay1 - ay0;
            const float acx = ax0 + 0.5f * aw, acy = ay0 + 0.5f * ah;
            const float gw = fmaxf(gx1 - gx0, 1.0f);
            const float gh = fmaxf(gy1 - gy0, 1.0f);
            const float gcx = gx0 + 0.5f * gw, gcy = gy0 + 0.5f * gh;
            const float t0 = ((gcx - acx) / aw) * 10.0f;   // / 0.1
            const float t1 = ((gcy - acy) / ah) * 10.0f;
            const float t2 = logf(gw / aw) * 5.0f;         // / 0.2
            const float t3 = logf(gh / ah) * 5.0f;
            const float r0 = reg_heads[base + 0];
            const float r1 = reg_heads[base + 1];
            const float r2 = reg_heads[base + 2];
            const float r3 = reg_heads[base + 3];
            float sl1 = 0.0f;
            const float xs[4] = { fabsf(r0 - t0), fabsf(r1 - t1),
                                  fabsf(r2 - t2), fabsf(r3 - t3) };
            #pragma unroll
            for (int k = 0; k < 4; ++k) {
                const float x = xs[k];
                sl1 += (x < BETA_F) ? (0.5f * x * x / BETA_F) : (x - 0.5f * BETA_F);
            }
            sl1pos = sl1 * 0.25f;
        }
    }

    // Every lane participates (WMMA needs full EXEC).
    block_atomic_add(sl1pos, &reg_sum[b], red);
    block_atomic_add(posf,   &num_pos[b], red);
}

// ---------------------------------------------------------------------------
// Kernel 2: streaming focal-loss reduction over cls_heads (float4 vectorized).
// grid = (1024, B), block = 256.
// ---------------------------------------------------------------------------
__device__ __forceinline__ float focal_term(float p, bool is_pos) {
    p = fminf(fmaxf(p, EPS_F), 1.0f - EPS_F);
    if (is_pos) {
        const float q = 1.0f - p;
        return ALPHA_F * q * q * (-__logf(p));
    } else {
        return (1.0f - ALPHA_F) * p * p * (-__logf(1.0f - p));
    }
}

__global__ void k_focal(const float* __restrict__ cls,
                        const float* __restrict__ labels,
                        float* __restrict__ cls_sum,
                        int N) {
    const int b = blockIdx.y;
    const float4* p4 = (const float4*)(cls + (size_t)b * N * NCLS);
    const float*  lb = labels + (size_t)b * N;
    const int total4 = N * (NCLS / 4);          // float4s per image
    const int vecsPerRow = NCLS / 4;            // 20

    __shared__ float red[8];

    float acc = 0.0f;
    for (int i = blockIdx.x * blockDim.x + threadIdx.x;
         i < total4;
         i += gridDim.x * blockDim.x) {
        const float4 p = p4[i];
        const int n  = i / vecsPerRow;
        const int c0 = (i - n * vecsPerRow) * 4;
        const float lab = lb[n];                // broadcast via L2/WGP$
        if (lab >= 0.0f) {
            const int pc = (int)lab - 1;        // -1 when lab==0 (background)
            acc += focal_term(p.x, c0 + 0 == pc);
            acc += focal_term(p.y, c0 + 1 == pc);
            acc += focal_term(p.z, c0 + 2 == pc);
            acc += focal_term(p.w, c0 + 3 == pc);
        }
    }

    block_atomic_add(acc, &cls_sum[b], red);
}

// ---------------------------------------------------------------------------
// Kernel 3: combine per-image sums into the two output scalars.
// ---------------------------------------------------------------------------
__global__ void k_final(const float* __restrict__ acc,   // [cls_sum[4], reg_sum[4], num_pos[4]]
                        float* __restrict__ out) {
    if (threadIdx.x == 0) {
        float cl = 0.0f, rl = 0.0f, nv = 0.0f;
        #pragma unroll
        for (int b = 0; b < BATCH; ++b) {
            const float np = acc[8 + b];
            if (np > 0.0f) {
                const float denom = fmaxf(np, 1.0f);
                cl += acc[b]     / denom;
                rl += acc[4 + b] / denom;
                nv += 1.0f;
            }
        }
        const float nvd = fmaxf(nv, 1.0f);
        out[0] = cl / nvd;
        out[1] = rl / nvd;
    }
}

// ---------------------------------------------------------------------------
extern "C" void kernel_launch(void* const* d_in, const int* in_sizes, int n_in,
                              void* d_out, int out_size, void* d_ws, size_t ws_size,
                              hipStream_t stream) {
    const float* cls_heads = (const float*)d_in[0];
    const float* reg_heads = (const float*)d_in[1];
    const float* anchors   = (const float*)d_in[2];
    const float* annots    = (const float*)d_in[3];
    float* out = (float*)d_out;

    const int N = in_sizes[1] / (BATCH * 4);          // reg_heads = B*N*4
    const int M = in_sizes[3] / (BATCH * 5);          // annotations = B*M*5

    float* labels  = (float*)d_ws;                    // B*N floats
    float* acc     = labels + (size_t)BATCH * N;      // 12 floats: cls[4], reg[4], npos[4]
    float* cls_sum = acc + 0;
    float* reg_sum = acc + 4;
    float* num_pos = acc + 8;

    k_zero<<<1, 32, 0, stream>>>(acc);

    dim3 gMatch((N + 255) / 256, BATCH);
    k_match<<<gMatch, 256, 0, stream>>>(anchors, reg_heads, annots,
                                        labels, reg_sum, num_pos, N, M);

    const int total4 = N * (NCLS / 4);
    int gx = (total4 + 255) / 256;
    if (gx > 1024) gx = 1024;
    dim3 gFocal(gx, BATCH);
    k_focal<<<gFocal, 256, 0, stream>>>(cls_heads, labels, cls_sum, N);

    k_final<<<1, 32, 0, stream>>>(acc, out);
}